// SymmetricLoss_77506979823875
// MI455X (gfx1250) — compile-verified
//
#include <hip/hip_runtime.h>

// ext vector types for the CDNA5 WMMA f32 16x16x4 builtin:
// A (16x4 f32) = 2 VGPRs/lane, B (4x16 f32) = 2 VGPRs/lane, C/D (16x16 f32) = 8 VGPRs/lane
typedef __attribute__((ext_vector_type(2))) float     v2f;
typedef __attribute__((ext_vector_type(8))) float     v8f;
typedef __attribute__((ext_vector_type(2))) long long v2l;

// Exact f32 sum across all 32 lanes of a wave using V_WMMA_F32_16X16X4_F32.
// A0 = per-lane partial, A1 = 0, B = all-ones. Each lane's partial occupies
// exactly one A-matrix slot, so D[m][n] = (sum of A row m) independent of n.
// C/D layout: lane n (0-15) VGPR j holds D[j][n]; lane 16+n holds D[j+8][n].
// => lane L's sum of d[0..7] = rows 0-7 (L<16) or rows 8-15 (L>=16);
//    adding the lane^16 partner gives the full wave sum on every lane.
__device__ __forceinline__ float wave_sum32(float p) {
    v2f a; a.x = p;    a.y = 0.0f;
    v2f b; b.x = 1.0f; b.y = 1.0f;
    v8f c = {};
    v8f d = __builtin_amdgcn_wmma_f32_16x16x4_f32(
        /*neg_a=*/false, a, /*neg_b=*/false, b,
        /*c_mod=*/(short)0, c, /*reuse_a=*/false, /*reuse_b=*/false);
    float s = d[0] + d[1] + d[2] + d[3] + d[4] + d[5] + d[6] + d[7];
    s += __shfl_xor(s, 16, 32);   // combine rows 0-7 with rows 8-15
    return s;                     // full wave sum, broadcast to all lanes
}

// Deterministic block reduction: WMMA wave sums -> LDS -> thread 0 scalar sum.
// Valid result only on threadIdx.x == 0.
__device__ __forceinline__ float block_sum(float p, float* lds) {
    float w = wave_sum32(p);                 // all lanes active: EXEC all-1s for WMMA
    const int lane = threadIdx.x & 31;
    const int wid  = threadIdx.x >> 5;
    if (lane == 0) lds[wid] = w;
    __syncthreads();
    float t = 0.0f;
    if (threadIdx.x == 0) {
        const int nw = (int)(blockDim.x >> 5);
        for (int i = 0; i < nw; ++i) t += lds[i];
    }
    return t;
}

__global__ void __launch_bounds__(256)
pair_dist_partial(const float* __restrict__ vt,
                  const long long* __restrict__ mt,
                  float* __restrict__ ws, int n_pairs) {
    __shared__ float lds[32];
    float acc = 0.0f;
    const int stride = (int)(gridDim.x * blockDim.x);
    for (int i = (int)(blockIdx.x * blockDim.x + threadIdx.x); i < n_pairs; i += stride) {
        // 16B vector load of the int64 index pair; non-temporal: the 32MB index
        // stream is single-use — keep L2 (192MB) for the 48MB vertex table.
        v2l pr = __builtin_nontemporal_load(((const v2l*)mt) + i);
        const int ia = (int)pr.x;          // indices < 4M, fit in i32
        const int ib = (int)pr.y;
        const float* __restrict__ pf = vt + 3 * ia;   // gathers hit L2
        const float* __restrict__ pt = vt + 3 * ib;
        const float fx = pf[0], fy = pf[1], fz = pf[2];
        const float tx = pt[0], ty = pt[1], tz = pt[2];
        const float dx = tx + fx;          // reference uses + on x
        const float dy = ty - fy;
        const float dz = tz - fz;
        acc += sqrtf(dx * dx + dy * dy + dz * dz);
    }
    const float t = block_sum(acc, lds);
    if (threadIdx.x == 0) ws[blockIdx.x] = t;
}

__global__ void __launch_bounds__(256)
final_reduce(const float* __restrict__ ws, float* __restrict__ out,
             int nb, float inv_n) {
    __shared__ float lds[32];
    float acc = 0.0f;
    for (int i = (int)threadIdx.x; i < nb; i += (int)blockDim.x) acc += ws[i];
    const float t = block_sum(acc, lds);
    if (threadIdx.x == 0) out[0] = t * inv_n;
}

extern "C" void kernel_launch(void* const* d_in, const int* in_sizes, int n_in,
                              void* d_out, int out_size, void* d_ws, size_t ws_size,
                              hipStream_t stream) {
    const float*     vt = (const float*)d_in[0];       // (N_VERTS, 3) f32
    const long long* mt = (const long long*)d_in[1];   // (N_PAIRS, 2) i64
    const int n_pairs = in_sizes[1] / 2;

    float* out = (float*)d_out;
    float* ws  = (float*)d_ws;                         // 1024 f32 partials (4KB)

    constexpr int NB = 1024;   // 1024 blocks x 8 wave32 = plenty to hide gather latency
    constexpr int NT = 256;

    pair_dist_partial<<<NB, NT, 0, stream>>>(vt, mt, ws, n_pairs);
    final_reduce<<<1, NT, 0, stream>>>(ws, out, NB, 1.0f / (float)n_pairs);
}